// CentroidTripletLoss_52956946759819
// MI455X (gfx1250) — compile-verified
//
#include <hip/hip_runtime.h>
#include <math.h>

typedef __attribute__((ext_vector_type(2))) float v2f;
typedef __attribute__((ext_vector_type(8))) float v8f;

#define NUM_CLASSES 256
#define DIM 512
#define NSAMP 65536
#define MARGIN_F 0.3f
#define EPS_F 1e-12f

#define DCHUNK 64                      // dims per accumulation block
#define NDC (DIM / DCHUNK)             // 8 dim-chunks
#define NSC 16                         // sample chunks
#define SCHUNK (NSAMP / NSC)           // 4096 samples per chunk

// workspace layout (float units)
#define OFF_SUMS 0
#define OFF_CENT (OFF_SUMS + NUM_CLASSES * DIM)   // 131072
#define OFF_INVN (OFF_CENT + NUM_CLASSES * DIM)   // 262144
#define OFF_CNT  (OFF_INVN + NSAMP)               // 327680
#define OFF_LSUM (OFF_CNT + NUM_CLASSES)          // 327936
#define OFF_NEAR (OFF_LSUM + NUM_CLASSES)         // 328192 (ints)

__global__ void k_zero(float* __restrict__ p, int n) {
  int i = blockIdx.x * blockDim.x + threadIdx.x;
  if (i < n) p[i] = 0.0f;
}

// One wave (32 lanes) per sample: compute inverse L2 norm; lane 0 bumps class count.
__global__ __launch_bounds__(256) void k_norm(const float* __restrict__ emb,
                                              const int* __restrict__ labels,
                                              float* __restrict__ counts,
                                              float* __restrict__ invn) {
  const int wave = threadIdx.x >> 5;
  const int lane = threadIdx.x & 31;
  const int s = blockIdx.x * 8 + wave;
  const float* row = emb + (size_t)s * DIM;

  float ss = 0.0f;
#pragma unroll
  for (int q = 0; q < 4; ++q) {
    float4 v = *(const float4*)(row + q * 128 + lane * 4);
    ss += v.x * v.x + v.y * v.y + v.z * v.z + v.w * v.w;
  }
#pragma unroll
  for (int off = 16; off >= 1; off >>= 1) ss += __shfl_xor(ss, off, 32);
  const float inv = 1.0f / fmaxf(sqrtf(ss), EPS_F);

  if (lane == 0) {
    invn[s] = inv;
    atomicAdd(counts + labels[s], 1.0f);
  }
}

// LDS-aggregated centroid accumulation.
// grid = (dim-chunk 0..7, sample-chunk 0..15); block = 256 threads (8 waves).
// Each block owns a 256-class x 64-dim fp32 accumulator in LDS (64 KB, fits CDNA5's
// 320 KB/WGP). Waves stream one sample per iteration: lane holds 2 dims (float2,
// 256 B coalesced per wave), ds_add_f32 into LDS (banks 2l / 2l+1: conflict-free).
// Flush = 16K global atomics/block -> 2M total (vs 33.5M if done directly in global).
__global__ __launch_bounds__(256) void k_accum_lds(const float* __restrict__ emb,
                                                   const int* __restrict__ labels,
                                                   const float* __restrict__ invn,
                                                   float* __restrict__ sums) {
  __shared__ float acc[NUM_CLASSES * DCHUNK];  // 64 KB

  const int dimc = blockIdx.x;   // 0..NDC-1
  const int sc = blockIdx.y;     // 0..NSC-1
  const int wave = threadIdx.x >> 5;
  const int lane = threadIdx.x & 31;

  for (int i = threadIdx.x; i < NUM_CLASSES * DCHUNK; i += 256) acc[i] = 0.0f;
  __syncthreads();

  const int d0 = dimc * DCHUNK + lane * 2;
#pragma unroll 4
  for (int it = 0; it < SCHUNK / 8; ++it) {
    const int s = sc * SCHUNK + it * 8 + wave;
    const int lab = labels[s];
    const float inv = invn[s];
    float2 e = *(const float2*)(emb + (size_t)s * DIM + d0);
    atomicAdd(&acc[lab * DCHUNK + lane * 2 + 0], e.x * inv);
    atomicAdd(&acc[lab * DCHUNK + lane * 2 + 1], e.y * inv);
  }
  __syncthreads();

  for (int i = threadIdx.x; i < NUM_CLASSES * DCHUNK; i += 256) {
    const int cls = i >> 6;       // i / DCHUNK
    const int dd = i & (DCHUNK - 1);
    atomicAdd(&sums[(size_t)cls * DIM + dimc * DCHUNK + dd], acc[i]);
  }
}

// One wave per class: cent = normalize(sums / max(count,1))
__global__ __launch_bounds__(256) void k_cent(const float* __restrict__ sums,
                                              const float* __restrict__ counts,
                                              float* __restrict__ cent) {
  const int wave = threadIdx.x >> 5;
  const int lane = threadIdx.x & 31;
  const int c = blockIdx.x * 8 + wave;
  const float invc = 1.0f / fmaxf(counts[c], 1.0f);
  const float* src = sums + (size_t)c * DIM;
  float4 v[4];
  float ss = 0.0f;
#pragma unroll
  for (int q = 0; q < 4; ++q) {
    float4 t = *(const float4*)(src + q * 128 + lane * 4);
    t.x *= invc; t.y *= invc; t.z *= invc; t.w *= invc;
    v[q] = t;
    ss += t.x * t.x + t.y * t.y + t.z * t.z + t.w * t.w;
  }
#pragma unroll
  for (int off = 16; off >= 1; off >>= 1) ss += __shfl_xor(ss, off, 32);
  const float inv = 1.0f / fmaxf(sqrtf(ss), EPS_F);
  float* dst = cent + (size_t)c * DIM;
#pragma unroll
  for (int q = 0; q < 4; ++q) {
    const int k = q * 128 + lane * 4;
    float4 t = v[q];
    t.x *= inv; t.y *= inv; t.z *= inv; t.w *= inv;
    *(float4*)(dst + k) = t;
  }
}

// Gram matrix via fp32 WMMA (exact fp32 like reference) + nearest-centroid argmin.
// Block = 16 waves; blockIdx = 16-row strip; wave w = 16-col tile.
__global__ __launch_bounds__(512) void k_gram_nearest(const float* __restrict__ cent,
                                                      const float* __restrict__ counts,
                                                      int* __restrict__ nearest) {
  __shared__ float lds_d[16][16];
  __shared__ int   lds_c[16][16];

  const int lane = threadIdx.x & 31;
  const int wv = threadIdx.x >> 5;   // column tile 0..15
  const int i0 = blockIdx.x * 16;    // row base
  const int m = lane & 15;
  const int kh = (lane >> 4) * 2;    // K sub-offset per ISA A/B fragment layout

  const float* arow = cent + (size_t)(i0 + m) * DIM + kh;       // A: rows of strip
  const float* brow = cent + (size_t)(wv * 16 + m) * DIM + kh;  // B: rows of col tile (Gram)

  v8f acc = {};
#pragma unroll 8
  for (int k = 0; k < DIM; k += 4) {
    v2f a = *(const v2f*)(arow + k);
    v2f b = *(const v2f*)(brow + k);
    acc = __builtin_amdgcn_wmma_f32_16x16x4_f32(
        /*neg_a=*/false, a, /*neg_b=*/false, b,
        /*c_mod=*/(short)0, acc, /*reuse_a=*/false, /*reuse_b=*/false);
  }

  const int col = wv * 16 + m;
  const bool col_present = counts[col] > 0.0f;
  const float INF = __builtin_inff();

#pragma unroll
  for (int r = 0; r < 8; ++r) {
    const int lr = r + ((lane >> 4) << 3);  // local row 0..15
    const int row = i0 + lr;
    float g = acc[r];
    float d = sqrtf(fmaxf(2.0f - 2.0f * g, 0.0f));
    if (row == col || !col_present) d = INF;
    int bc = col;
    // min over the 16 columns of this tile; ties -> smallest col (argmin semantics)
#pragma unroll
    for (int off = 8; off >= 1; off >>= 1) {
      float od = __shfl_xor(d, off, 16);
      int oc = __shfl_xor(bc, off, 16);
      if (od < d || (od == d && oc < bc)) { d = od; bc = oc; }
    }
    if (m == 0) { lds_d[lr][wv] = d; lds_c[lr][wv] = bc; }
  }
  __syncthreads();

  if (threadIdx.x < 256) {
    const int lr = threadIdx.x >> 4;
    const int j = threadIdx.x & 15;
    float d = lds_d[lr][j];
    int bc = lds_c[lr][j];
#pragma unroll
    for (int off = 8; off >= 1; off >>= 1) {
      float od = __shfl_xor(d, off, 16);
      int oc = __shfl_xor(bc, off, 16);
      if (od < d || (od == d && oc < bc)) { d = od; bc = oc; }
    }
    if (j == 0) nearest[i0 + lr] = bc;
  }
}

// One wave per sample: triplet term, accumulate per class.
__global__ __launch_bounds__(256) void k_loss(const float* __restrict__ emb,
                                              const int* __restrict__ labels,
                                              const float* __restrict__ cent,
                                              const int* __restrict__ nearest,
                                              const float* __restrict__ invn,
                                              float* __restrict__ lsum) {
  const int wave = threadIdx.x >> 5;
  const int lane = threadIdx.x & 31;
  const int s = blockIdx.x * 8 + wave;
  const int lab = labels[s];
  const int neg = nearest[lab];
  const float inv = invn[s];
  const float* row = emb + (size_t)s * DIM;
  const float* pc = cent + (size_t)lab * DIM;
  const float* nc = cent + (size_t)neg * DIM;

  float dp = 0.0f, dn = 0.0f;
#pragma unroll
  for (int q = 0; q < 4; ++q) {
    const int k = q * 128 + lane * 4;
    float4 e = *(const float4*)(row + k);
    float4 p = *(const float4*)(pc + k);
    float4 n = *(const float4*)(nc + k);
    dp += e.x * p.x + e.y * p.y + e.z * p.z + e.w * p.w;
    dn += e.x * n.x + e.y * n.y + e.z * n.z + e.w * n.w;
  }
  dp *= inv;
  dn *= inv;
#pragma unroll
  for (int off = 16; off >= 1; off >>= 1) {
    dp += __shfl_xor(dp, off, 32);
    dn += __shfl_xor(dn, off, 32);
  }
  if (lane == 0) {
    // d_pos - d_neg + margin = (1-dp) - (1-dn) + m = dn - dp + m
    const float term = fmaxf(dn - dp + MARGIN_F, 0.0f);
    atomicAdd(lsum + lab, term);
  }
}

__global__ __launch_bounds__(256) void k_final(const float* __restrict__ counts,
                                               const float* __restrict__ lsum,
                                               float* __restrict__ out) {
  __shared__ float sv[8];
  __shared__ float sp[8];
  const int t = threadIdx.x;
  const float cnt = counts[t];
  const bool present = cnt > 0.0f;
  const float lc = lsum[t] / fmaxf(cnt, 1.0f);
  float val = present ? lc : 0.0f;
  float pc = present ? 1.0f : 0.0f;
#pragma unroll
  for (int off = 16; off >= 1; off >>= 1) {
    val += __shfl_xor(val, off, 32);
    pc += __shfl_xor(pc, off, 32);
  }
  if ((t & 31) == 0) { sv[t >> 5] = val; sp[t >> 5] = pc; }
  __syncthreads();
  if (t == 0) {
    float a = 0.0f, b = 0.0f;
#pragma unroll
    for (int i = 0; i < 8; ++i) { a += sv[i]; b += sp[i]; }
    out[0] = a / fmaxf(b, 1.0f);
  }
}

extern "C" void kernel_launch(void* const* d_in, const int* in_sizes, int n_in,
                              void* d_out, int out_size, void* d_ws, size_t ws_size,
                              hipStream_t stream) {
  (void)in_sizes; (void)n_in; (void)out_size; (void)ws_size;
  const float* emb = (const float*)d_in[0];
  const int* labels = (const int*)d_in[1];
  float* ws = (float*)d_ws;
  float* sums = ws + OFF_SUMS;
  float* cent = ws + OFF_CENT;
  float* invn = ws + OFF_INVN;
  float* counts = ws + OFF_CNT;
  float* lsum = ws + OFF_LSUM;
  int* nearest = (int*)(ws + OFF_NEAR);
  float* out = (float*)d_out;

  // zero accumulators (sums; counts+lsum are contiguous)
  k_zero<<<(NUM_CLASSES * DIM + 511) / 512, 512, 0, stream>>>(sums, NUM_CLASSES * DIM);
  k_zero<<<1, 512, 0, stream>>>(counts, 2 * NUM_CLASSES);

  k_norm<<<NSAMP / 8, 256, 0, stream>>>(emb, labels, counts, invn);
  {
    dim3 grid(NDC, NSC);
    k_accum_lds<<<grid, 256, 0, stream>>>(emb, labels, invn, sums);
  }
  k_cent<<<NUM_CLASSES / 8, 256, 0, stream>>>(sums, counts, cent);
  k_gram_nearest<<<NUM_CLASSES / 16, 512, 0, stream>>>(cent, counts, nearest);
  k_loss<<<NSAMP / 8, 256, 0, stream>>>(emb, labels, cent, nearest, invn, lsum);
  k_final<<<1, 256, 0, stream>>>(counts, lsum, out);
}